// AttnDecoderRNN_full_33165737459961
// MI455X (gfx1250) — compile-verified
//
#include <hip/hip_runtime.h>
#include <math.h>

// ---------------- problem constants ----------------
#define H    512
#define H2   1024      // 2H
#define H4   2048      // 4H
#define E    256
#define XIN  1280      // 2H+E
#define V3   50003
#define TK   4096
#define PGN  2304      // 4H+E

// d_out layout (reference tuple, concatenated flat, fp32):
#define OUT_FINAL 0                   // [V3]
#define OUT_H     (OUT_FINAL + V3)    // [H]
#define OUT_C     (OUT_H + H)         // [H]
#define OUT_CT    (OUT_C + H)         // [2H]
#define OUT_ATTN  (OUT_CT + H2)       // [TK]
#define OUT_PGEN  (OUT_ATTN + TK)     // [1]
#define OUT_COV   (OUT_PGEN + 1)      // [TK]

// workspace layout (floats)
#define WS_X       0
#define WS_GATES   (WS_X + E)          // 2048
#define WS_DECHAT  (WS_GATES + H4)     // 1024  = [h_new, c_new]
#define WS_DECFEAT (WS_DECHAT + H2)    // 1024
#define WS_SCORES  (WS_DECFEAT + H2)   // 4096
#define WS_ATTN    (WS_SCORES + TK)    // 4096
#define WS_OUTH    (WS_ATTN + TK)      // 512
#define WS_PGEN    (WS_OUTH + H)       // 1
#define WS_LOGITS  (WS_PGEN + 8)       // 50003   (needs ~253KB total)

typedef __attribute__((ext_vector_type(2))) float v2f;
typedef __attribute__((ext_vector_type(8))) float v8f;

__device__ __forceinline__ float sigmoidf_(float x) { return 1.0f / (1.0f + __expf(-x)); }

// ---- 1) x = W_xc @ concat(c_t_1, emb[decoder_input]) + b_xc -----------------
__global__ void k_xcontext(const float* __restrict__ c_t_1,
                           const float* __restrict__ emb,
                           const float* __restrict__ W_xc,
                           const float* __restrict__ b_xc,
                           const int*   __restrict__ dec_in,
                           float* __restrict__ x) {
  int j = threadIdx.x;                       // 256 threads
  const float* erow = emb + (size_t)dec_in[0] * E;
  const float* wrow = W_xc + (size_t)j * XIN;
  float s = b_xc[j];
  for (int k = 0; k < H2; ++k)  s = fmaf(wrow[k], c_t_1[k], s);
  for (int k = 0; k < E;  ++k)  s = fmaf(wrow[H2 + k], erow[k], s);
  x[j] = s;
}

// ---- 2) gates = W_ih@x + W_hh@h0 + b_ih + b_hh ------------------------------
__global__ void k_gates(const float* __restrict__ x,
                        const float* __restrict__ h0,
                        const float* __restrict__ W_ih,
                        const float* __restrict__ W_hh,
                        const float* __restrict__ b_ih,
                        const float* __restrict__ b_hh,
                        float* __restrict__ gates) {
  int j = blockIdx.x * blockDim.x + threadIdx.x;   // 2048
  const float* wi = W_ih + (size_t)j * E;
  const float* wh = W_hh + (size_t)j * H;
  float s = b_ih[j] + b_hh[j];
  for (int k = 0; k < E; ++k) s = fmaf(wi[k], x[k], s);
  for (int k = 0; k < H; ++k) s = fmaf(wh[k], h0[k], s);
  gates[j] = s;
}

// ---- 3) LSTM cell: c_new, h_new, dec_hat=[h,c] ------------------------------
__global__ void k_lstm(const float* __restrict__ gates,
                       const float* __restrict__ c0,
                       float* __restrict__ out,
                       float* __restrict__ dec_hat) {
  int j = threadIdx.x;                       // 512 threads
  float i = gates[j], f = gates[H + j], g = gates[2 * H + j], o = gates[3 * H + j];
  float c = sigmoidf_(f) * c0[j] + sigmoidf_(i) * tanhf(g);
  float h = sigmoidf_(o) * tanhf(c);
  out[OUT_H + j] = h;
  out[OUT_C + j] = c;
  dec_hat[j]     = h;
  dec_hat[H + j] = c;
}

// ---- 4) dec_feat = W_dp @ dec_hat + b_dp ------------------------------------
__global__ void k_decfeat(const float* __restrict__ dec_hat,
                          const float* __restrict__ W_dp,
                          const float* __restrict__ b_dp,
                          float* __restrict__ dec_feat) {
  int j = blockIdx.x * blockDim.x + threadIdx.x;   // 1024
  const float* w = W_dp + (size_t)j * H2;
  float s = b_dp[j];
  for (int k = 0; k < H2; ++k) s = fmaf(w[k], dec_hat[k], s);
  dec_feat[j] = s;
}

// ---- 5) WMMA attention scores ----------------------------------------------
// One block = 32 timesteps (two 16-row A tiles staged in LDS via the gfx1250
// async global->LDS path). 16 waves; each wave owns 4 column-tiles of W_h.
// Per K-step one B fragment feeds TWO chained v_wmma_f32_16x16x4_f32 (one per
// row tile), halving B traffic vs 16-row blocks. tanh/v_att epilogue reduces
// per-timestep scores via shfl + LDS atomics.
__global__ void __launch_bounds__(512)
k_attn_scores(const float* __restrict__ enc,
              const float* __restrict__ W_h,
              const float* __restrict__ dec_feat,
              const float* __restrict__ coverage,
              const float* __restrict__ w_c,
              const float* __restrict__ v_att,
              float* __restrict__ scores) {
  extern __shared__ float smem[];              // 32*1024 enc tile + 32 scores
  float* s_enc   = smem;
  float* s_score = smem + 32 * H2;

  const int tid = threadIdx.x;
  const int t0  = blockIdx.x * 32;

  // async stage of contiguous 32x1024 f32 tile: global -> LDS, no VGPR bounce
  {
    const uint64_t gbase = (uint64_t)(uintptr_t)(enc + (size_t)t0 * H2);
    const uint32_t lbase = (uint32_t)(uintptr_t)s_enc;   // LDS byte offset
    for (int i = tid; i < 32 * H2 / 4; i += 512) {
      uint32_t loff = lbase + (uint32_t)i * 16u;
      uint64_t ga   = gbase + (uint64_t)i * 16u;
      asm volatile("global_load_async_to_lds_b128 %0, %1, off"
                   :: "v"(loff), "v"(ga) : "memory");
    }
  }
  if (tid < 32) s_score[tid] = 0.0f;
  asm volatile("s_wait_asynccnt 0x0" ::: "memory");
  __syncthreads();

  const int wv    = tid >> 5;
  const int lane  = tid & 31;
  const int mrow  = lane & 15;        // A: M row in tile / B: N col in tile
  const int koff  = (lane >> 4) * 2;  // K sub-offset per ISA 16x4 f32 layout
  const int thalf = (lane >> 4) * 8;

  const float* arow0 = s_enc + (size_t)mrow * H2 + koff;         // rows 0..15
  const float* arow1 = s_enc + (size_t)(16 + mrow) * H2 + koff;  // rows 16..31

  for (int mt = wv * 4; mt < wv * 4 + 4; ++mt) {
    const float* brow = W_h + (size_t)(mt * 16 + mrow) * H2 + koff;
    v8f c0 = {0.f, 0.f, 0.f, 0.f, 0.f, 0.f, 0.f, 0.f};
    v8f c1 = {0.f, 0.f, 0.f, 0.f, 0.f, 0.f, 0.f, 0.f};
#pragma unroll 4
    for (int kb = 0; kb < H2; kb += 4) {
      v2f b  = *reinterpret_cast<const v2f*>(brow + kb);    // global_load_b64
      v2f a0 = *reinterpret_cast<const v2f*>(arow0 + kb);   // ds_load_b64
      v2f a1 = *reinterpret_cast<const v2f*>(arow1 + kb);
      c0 = __builtin_amdgcn_wmma_f32_16x16x4_f32(
               false, a0, false, b, (short)0, c0, false, false);
      c1 = __builtin_amdgcn_wmma_f32_16x16x4_f32(
               false, a1, false, b, (short)0, c1, false, false);
    }
    // epilogue: tanh(enc_feat + dec_feat + cov*w_c) * v_att, reduce over N
    const int n  = mt * 16 + mrow;
    const float dn = dec_feat[n], wcn = w_c[n], vn = v_att[n];
#pragma unroll
    for (int r = 0; r < 8; ++r) {
      const int ta = t0 + thalf + r;
      const int tb = ta + 16;
      float v0 = tanhf(c0[r] + dn + coverage[ta] * wcn) * vn;
      float v1 = tanhf(c1[r] + dn + coverage[tb] * wcn) * vn;
      for (int m = 8; m >= 1; m >>= 1) {
        v0 += __shfl_xor(v0, m, 16);
        v1 += __shfl_xor(v1, m, 16);
      }
      if (mrow == 0) {
        atomicAdd(&s_score[thalf + r],      v0);
        atomicAdd(&s_score[16 + thalf + r], v1);
      }
    }
  }
  __syncthreads();
  if (tid < 32) scores[t0 + tid] = s_score[tid];
}

// ---- 6) softmax over TK, renormalize, attn + coverage_next ------------------
//        (also zero-fills the c_t output region for k_context's atomics)
__global__ void k_softmax_attn(const float* __restrict__ scores,
                               const float* __restrict__ coverage,
                               float* __restrict__ attn_ws,
                               float* __restrict__ out) {
  __shared__ float red[1024];
  const int tid = threadIdx.x;
  out[OUT_CT + tid] = 0.0f;                      // 1024 threads == H2
  float lm = -INFINITY;
  for (int t = tid; t < TK; t += 1024) lm = fmaxf(lm, scores[t]);
  red[tid] = lm; __syncthreads();
  for (int s = 512; s; s >>= 1) { if (tid < s) red[tid] = fmaxf(red[tid], red[tid + s]); __syncthreads(); }
  const float mx = red[0]; __syncthreads();

  float ls = 0.f;
  for (int t = tid; t < TK; t += 1024) ls += __expf(scores[t] - mx);
  red[tid] = ls; __syncthreads();
  for (int s = 512; s; s >>= 1) { if (tid < s) red[tid] += red[tid + s]; __syncthreads(); }
  const float sum = red[0]; __syncthreads();

  // reference renormalizes attn by its own sum
  float la = 0.f;
  for (int t = tid; t < TK; t += 1024) la += __expf(scores[t] - mx) / sum;
  red[tid] = la; __syncthreads();
  for (int s = 512; s; s >>= 1) { if (tid < s) red[tid] += red[tid + s]; __syncthreads(); }
  const float inv = 1.0f / (sum * red[0]);

  for (int t = tid; t < TK; t += 1024) {
    float a = __expf(scores[t] - mx) * inv;
    attn_ws[t]        = a;
    out[OUT_ATTN + t] = a;
    out[OUT_COV  + t] = coverage[t] + a;
  }
}

// ---- 7) c_t = attn @ enc : 64 timestep-chunks x 4 column-blocks -------------
__global__ void k_context(const float* __restrict__ attn,
                          const float* __restrict__ enc,
                          float* __restrict__ out) {
  int n  = blockIdx.x * blockDim.x + threadIdx.x;  // column, coalesced
  int tb = blockIdx.y * (TK / 64);                 // 64-timestep chunk
  float s = 0.f;
  for (int t = tb; t < tb + TK / 64; ++t)
    s = fmaf(attn[t], enc[(size_t)t * H2 + n], s);
  unsafeAtomicAdd(&out[OUT_CT + n], s);            // global_atomic_add_f32
}

// ---- 8) p_gen ----------------------------------------------------------------
__global__ void k_pgen(const float* __restrict__ out_ct,   // d_out + OUT_CT
                       const float* __restrict__ dec_hat,
                       const float* __restrict__ x,
                       const float* __restrict__ W_pg,
                       const float* __restrict__ b_pg,
                       float* __restrict__ out,
                       float* __restrict__ pgen_ws) {
  __shared__ float red[256];
  const int tid = threadIdx.x;
  float s = 0.f;
  for (int k = tid; k < PGN; k += 256) {
    float in = (k < H2) ? out_ct[k] : (k < 2 * H2) ? dec_hat[k - H2] : x[k - 2 * H2];
    s = fmaf(W_pg[k], in, s);
  }
  red[tid] = s; __syncthreads();
  for (int st = 128; st; st >>= 1) { if (tid < st) red[tid] += red[tid + st]; __syncthreads(); }
  if (tid == 0) {
    float p = sigmoidf_(red[0] + b_pg[0]);
    out[OUT_PGEN] = p;
    pgen_ws[0] = p;
  }
}

// ---- 9) out = W_o1 @ concat(h_new, c_t) + b_o1 ------------------------------
__global__ void k_out1(const float* __restrict__ dout,   // d_out base
                       const float* __restrict__ W_o1,
                       const float* __restrict__ b_o1,
                       float* __restrict__ outh) {
  int j = blockIdx.x * blockDim.x + threadIdx.x;   // 512
  const float* w = W_o1 + (size_t)j * (3 * H);
  float s = b_o1[j];
  for (int k = 0; k < H; ++k)  s = fmaf(w[k], dout[OUT_H + k], s);
  for (int k = 0; k < H2; ++k) s = fmaf(w[H + k], dout[OUT_CT + k], s);
  outh[j] = s;
}

// ---- 10) logits = W_o2 @ out + b_o2 : one wave32 per row, b128 loads --------
__global__ void k_logits(const float* __restrict__ W_o2,
                         const float* __restrict__ b_o2,
                         const float* __restrict__ outh,
                         float* __restrict__ logits) {
  const int lane = threadIdx.x & 31;
  const int v = blockIdx.x * 8 + (threadIdx.x >> 5);
  if (v >= V3) return;
  const float4* w4 = reinterpret_cast<const float4*>(W_o2 + (size_t)v * H);
  const float4* o4 = reinterpret_cast<const float4*>(outh);
  float s = 0.f;
#pragma unroll
  for (int i = 0; i < H / 128; ++i) {              // 4 x global_load_b128/lane
    float4 w = w4[lane + 32 * i];
    float4 o = o4[lane + 32 * i];
    s = fmaf(w.x, o.x, s); s = fmaf(w.y, o.y, s);
    s = fmaf(w.z, o.z, s); s = fmaf(w.w, o.w, s);
  }
  for (int m = 16; m >= 1; m >>= 1) s += __shfl_xor(s, m, 32);
  if (lane == 0) logits[v] = s + b_o2[v];
}

// ---- 11) vocab softmax -> final_dist = p_gen * softmax(logits) --------------
__global__ void k_vocab_softmax(const float* __restrict__ logits,
                                const float* __restrict__ pgen_ws,
                                float* __restrict__ out) {
  __shared__ float red[1024];
  const int tid = threadIdx.x;
  float lm = -INFINITY;
  for (int v = tid; v < V3; v += 1024) lm = fmaxf(lm, logits[v]);
  red[tid] = lm; __syncthreads();
  for (int s = 512; s; s >>= 1) { if (tid < s) red[tid] = fmaxf(red[tid], red[tid + s]); __syncthreads(); }
  const float mx = red[0]; __syncthreads();

  float ls = 0.f;
  for (int v = tid; v < V3; v += 1024) ls += __expf(logits[v] - mx);
  red[tid] = ls; __syncthreads();
  for (int s = 512; s; s >>= 1) { if (tid < s) red[tid] += red[tid + s]; __syncthreads(); }
  const float scale = pgen_ws[0] / red[0];

  for (int v = tid; v < V3; v += 1024)
    out[OUT_FINAL + v] = __expf(logits[v] - mx) * scale;
}

// ---- 12) scatter-add (1-p_gen)*attn at input_idx ----------------------------
__global__ void k_scatter(const float* __restrict__ attn,
                          const int*   __restrict__ input_idx,
                          const float* __restrict__ pgen_ws,
                          float* __restrict__ out) {
  int t = blockIdx.x * blockDim.x + threadIdx.x;   // 4096
  float w = (1.0f - pgen_ws[0]) * attn[t];
  unsafeAtomicAdd(&out[OUT_FINAL + input_idx[t]], w);
}

// ============================ launcher =======================================
extern "C" void kernel_launch(void* const* d_in, const int* in_sizes, int n_in,
                              void* d_out, int out_size, void* d_ws, size_t ws_size,
                              hipStream_t stream) {
  const float* enc      = (const float*)d_in[0];
  const float* c_t_1    = (const float*)d_in[1];
  const float* h0       = (const float*)d_in[2];
  const float* c0       = (const float*)d_in[3];
  const float* coverage = (const float*)d_in[4];
  const float* emb      = (const float*)d_in[5];
  const float* W_xc     = (const float*)d_in[6];
  const float* b_xc     = (const float*)d_in[7];
  const float* W_ih     = (const float*)d_in[8];
  const float* W_hh     = (const float*)d_in[9];
  const float* b_ih     = (const float*)d_in[10];
  const float* b_hh     = (const float*)d_in[11];
  const float* W_h      = (const float*)d_in[12];
  const float* W_dp     = (const float*)d_in[13];
  const float* b_dp     = (const float*)d_in[14];
  const float* w_c      = (const float*)d_in[15];
  const float* v_att    = (const float*)d_in[16];
  const float* W_pg     = (const float*)d_in[17];
  const float* b_pg     = (const float*)d_in[18];
  const float* W_o1     = (const float*)d_in[19];
  const float* b_o1     = (const float*)d_in[20];
  const float* W_o2     = (const float*)d_in[21];
  const float* b_o2     = (const float*)d_in[22];
  const int*   dec_in   = (const int*)d_in[23];
  const int*   in_idx   = (const int*)d_in[24];

  float* out = (float*)d_out;
  float* ws  = (float*)d_ws;

  k_xcontext<<<1, 256, 0, stream>>>(c_t_1, emb, W_xc, b_xc, dec_in, ws + WS_X);
  k_gates<<<8, 256, 0, stream>>>(ws + WS_X, h0, W_ih, W_hh, b_ih, b_hh, ws + WS_GATES);
  k_lstm<<<1, 512, 0, stream>>>(ws + WS_GATES, c0, out, ws + WS_DECHAT);
  k_decfeat<<<4, 256, 0, stream>>>(ws + WS_DECHAT, W_dp, b_dp, ws + WS_DECFEAT);

  const size_t smem = (32 * H2 + 32) * sizeof(float);   // 131,200 B dynamic LDS
  k_attn_scores<<<TK / 32, 512, smem, stream>>>(
      enc, W_h, ws + WS_DECFEAT, coverage, w_c, v_att, ws + WS_SCORES);

  k_softmax_attn<<<1, 1024, 0, stream>>>(ws + WS_SCORES, coverage, ws + WS_ATTN, out);
  k_context<<<dim3(H2 / 256, 64), 256, 0, stream>>>(ws + WS_ATTN, enc, out);
  k_pgen<<<1, 256, 0, stream>>>(out + OUT_CT, ws + WS_DECHAT, ws + WS_X,
                                W_pg, b_pg, out, ws + WS_PGEN);
  k_out1<<<2, 256, 0, stream>>>(out, W_o1, b_o1, ws + WS_OUTH);
  k_logits<<<(V3 + 7) / 8, 256, 0, stream>>>(W_o2, b_o2, ws + WS_OUTH, ws + WS_LOGITS);
  k_vocab_softmax<<<1, 1024, 0, stream>>>(ws + WS_LOGITS, ws + WS_PGEN, out);
  k_scatter<<<TK / 256, 256, 0, stream>>>(ws + WS_ATTN, in_idx, ws + WS_PGEN, out);
}